// RNN_32950989095215
// MI455X (gfx1250) — compile-verified
//
#include <hip/hip_runtime.h>

// ---------------------------------------------------------------------------
// RNN forward for MI455X (gfx1250, wave32).
//   h_{t} = tanh(x_t @ Wx^T + b + h_{t-1} @ Wh^T);  out = mean_t(h) @ Wo^T + bo
//
// 16 workgroups (one per 16-batch-row tile), 512 threads = 16 waves each.
// Wave w computes output columns [16w, 16w+16) using V_WMMA_F32_16X16X4_F32,
// with its Wh slice held entirely in VGPRs (64 x v2f = 128 VGPRs/lane).
// Hidden state double-buffered in LDS (stride 260 -> conflict-free b64 loads).
// ---------------------------------------------------------------------------

typedef __attribute__((ext_vector_type(2))) float v2f;
typedef __attribute__((ext_vector_type(4))) float v4f;
typedef __attribute__((ext_vector_type(8))) float v8f;

#define T_STEPS 1024
#define HID     256
#define BTILE   16
#define NWAVES  16
#define CHUNK   64      // timesteps of x staged in LDS per reload
#define HP      260     // padded row stride in floats (260*4B: 4-bank stagger)

__global__ __launch_bounds__(512)
void rnn_scan_kernel(const float* __restrict__ x,       // (256,1024,4)
                     const float* __restrict__ W_i2h,   // (256,260) = [Wx | Wh]
                     const float* __restrict__ b_i2h,   // (256)
                     const float* __restrict__ W_h2o,   // (1,256)
                     const float* __restrict__ b_h2o,   // (1)
                     const float* __restrict__ h0v,     // (256)
                     float* __restrict__ out)           // (256)
{
    __shared__ float h_lds[2][BTILE * HP];   // double-buffered hidden tile
    __shared__ float xs[BTILE * HP];         // x chunk: [m][tt*4+i], tt<64
    __shared__ float red[BTILE];             // per-batch-row output reduction

    const int tid  = threadIdx.x;
    const int lane = tid & 31;
    const int wave = tid >> 5;          // 0..15  -> N tile index
    const int half = lane >> 4;         // 0/1    -> K-pair select (A/B frags)
    const int l16  = lane & 15;         // A: row M;  B/C/D: column within tile
    const int b0   = blockIdx.x * BTILE;
    const int n    = wave * 16 + l16;   // this lane's global hidden column

    // ---- per-lane constants -------------------------------------------------
    const v4f  wx   = *(const v4f*)(W_i2h + (size_t)n * 260);      // Wx[n][0..3]
    const float bias = b_i2h[n];
    const float wout = W_h2o[n];

    // ---- Wh slice into registers: bw[kk] = Wh[n][4kk+2*half .. +1] ----------
    v2f bw[64];
    {
        const float* whrow = W_i2h + (size_t)n * 260 + 4 + 2 * half;
        #pragma unroll
        for (int kk = 0; kk < 64; ++kk)
            bw[kk] = *(const v2f*)(whrow + 4 * kk);
    }

    // ---- init hidden state & reduction buffer -------------------------------
    for (int idx = tid; idx < BTILE * HID; idx += 512) {
        int m = idx >> 8, c = idx & 255;
        h_lds[0][m * HP + c] = h0v[c];
    }
    if (tid < BTILE) red[tid] = 0.0f;

    v8f pooled = {};                    // sum over t of h for this lane's elems
    int buf = 0;

    for (int tc = 0; tc < T_STEPS / CHUNK; ++tc) {
        // stage x chunk: rows b0..b0+15, CHUNK*4 = 256 contiguous floats each
        for (int idx = tid; idx < BTILE * CHUNK; idx += 512) {
            int m = idx >> 6, c4 = idx & 63;
            v4f v = *(const v4f*)(x + (size_t)(b0 + m) * (T_STEPS * 4)
                                    + tc * (CHUNK * 4) + c4 * 4);
            *(v4f*)(xs + m * HP + c4 * 4) = v;
        }
        __syncthreads();

        for (int tt = 0; tt < CHUNK; ++tt) {
            // ---- accumulator init: xproj = x_t . Wx^T + b (K=4, folded) -----
            v8f acc;
            #pragma unroll
            for (int r = 0; r < 8; ++r) {
                int m = r + 8 * half;                  // C/D row for this lane
                v4f xv = *(const v4f*)(xs + m * HP + tt * 4);
                acc[r] = bias + xv.x * wx.x + xv.y * wx.y
                              + xv.z * wx.z + xv.w * wx.w;
            }
            v8f c1 = {}, c2 = {}, c3 = {};

            // ---- h_prev @ Wh^T : 64 x V_WMMA_F32_16X16X4_F32, 4 chains ------
            const float* hrow = &h_lds[buf][l16 * HP + 2 * half];
            #pragma unroll
            for (int kk = 0; kk < 64; kk += 4) {
                v2f a0 = *(const v2f*)(hrow + 4 * (kk + 0));
                acc = __builtin_amdgcn_wmma_f32_16x16x4_f32(
                        false, a0, false, bw[kk + 0], (short)0, acc, false, false);
                v2f a1 = *(const v2f*)(hrow + 4 * (kk + 1));
                c1  = __builtin_amdgcn_wmma_f32_16x16x4_f32(
                        false, a1, false, bw[kk + 1], (short)0, c1, false, false);
                v2f a2 = *(const v2f*)(hrow + 4 * (kk + 2));
                c2  = __builtin_amdgcn_wmma_f32_16x16x4_f32(
                        false, a2, false, bw[kk + 2], (short)0, c2, false, false);
                v2f a3 = *(const v2f*)(hrow + 4 * (kk + 3));
                c3  = __builtin_amdgcn_wmma_f32_16x16x4_f32(
                        false, a3, false, bw[kk + 3], (short)0, c3, false, false);
            }

            // ---- tanh, pool, publish h_new ----------------------------------
            #pragma unroll
            for (int r = 0; r < 8; ++r) {
                float v  = acc[r] + c1[r] + c2[r] + c3[r];
                float e  = __expf(2.0f * v);           // v_exp_f32
                float hn = 1.0f - 2.0f / (e + 1.0f);   // tanh(v)
                pooled[r] += hn;
                h_lds[buf ^ 1][(r + 8 * half) * HP + n] = hn;
            }
            buf ^= 1;
            __syncthreads();            // h_new visible; old buffer reusable
        }
    }

    // ---- out[b] = (1/T) * sum_n pooled[b][n] * Wo[n] + bo -------------------
    #pragma unroll
    for (int r = 0; r < 8; ++r)
        atomicAdd(&red[r + 8 * half], pooled[r] * wout);   // ds_add_f32
    __syncthreads();
    if (tid < BTILE)
        out[b0 + tid] = red[tid] * (1.0f / T_STEPS) + b_h2o[0];
}

extern "C" void kernel_launch(void* const* d_in, const int* in_sizes, int n_in,
                              void* d_out, int out_size, void* d_ws, size_t ws_size,
                              hipStream_t stream) {
    const float* x      = (const float*)d_in[0];
    const float* W_i2h  = (const float*)d_in[1];
    const float* b_i2h  = (const float*)d_in[2];
    const float* W_h2o  = (const float*)d_in[3];
    const float* b_h2o  = (const float*)d_in[4];
    const float* h0     = (const float*)d_in[5];
    float* out          = (float*)d_out;

    rnn_scan_kernel<<<256 / BTILE, 512, 0, stream>>>(
        x, W_i2h, b_i2h, W_h2o, b_h2o, h0, out);
}